// AttentionGate_55095840473688
// MI455X (gfx1250) — compile-verified
//
#include <hip/hip_runtime.h>

// CDNA5 / gfx1250, wave32. fp32 WMMA path: V_WMMA_F32_16X16X4_F32.
//
// Cache strategy (192 MB global L2, HBM 23.3 TB/s):
//   q (134 MB) fits in L2 and is read in BOTH pass 1 and pass 3 -> regular-temporal.
//   k (134 MB) is dead after pass 1                              -> non-temporal loads.
//   out (134 MB) is never re-read                                -> non-temporal stores.
// This keeps q resident across passes and targets ~400 MB of HBM traffic total.
typedef __attribute__((ext_vector_type(2))) float v2f;
typedef __attribute__((ext_vector_type(8))) float v8f;

#define CCH 32                      // channels
#define NSP (64 * 64 * 64)          // spatial N = 262144
#define K1_WGS 128                  // workgroups per batch, pass 1
#define K1_WAVES 8
#define K1_ROWS (NSP / (K1_WGS * K1_WAVES))   // 256 rows per wave
#define K3_WGS 512                  // workgroups per batch, pass 3
#define K3_ROWS (NSP / (K3_WGS * 8))          // 64 rows per wave

// -------- Pass 1: per-workgroup partial Gram w[i,j] = sum_n q[n,i]*k[n,j] ----
__global__ __launch_bounds__(256) void gram_kernel(const float* __restrict__ q,
                                                   const float* __restrict__ k,
                                                   float* __restrict__ partial) {
    const int wg   = blockIdx.x % K1_WGS;
    const int b    = blockIdx.x / K1_WGS;
    const int wave = threadIdx.x >> 5;
    const int lane = threadIdx.x & 31;
    const int lm   = lane & 15;
    const int kb   = (lane < 16) ? 0 : 2;   // A/B operand K sub-offset per half-wave

    const long n0 = (long)wg * (K1_WAVES * K1_ROWS) + (long)wave * K1_ROWS;
    const float* qb = q + ((long)b * NSP + n0) * CCH;
    const float* kp = k + ((long)b * NSP + n0) * CCH;

    v8f c00 = {}, c01 = {}, c10 = {}, c11 = {};
    for (int it = 0; it < K1_ROWS; it += 4) {
        const float* qr = qb + (long)(it + kb) * CCH;   // rows it+kb, it+kb+1
        const float* kr = kp + (long)(it + kb) * CCH;
        v2f a0, a1, b0, b1;
        a0.x = qr[lm];           a0.y = qr[CCH + lm];        // A[i=lm,      K]
        a1.x = qr[16 + lm];      a1.y = qr[CCH + 16 + lm];   // A[i=16+lm,   K]
        b0.x = __builtin_nontemporal_load(kr + lm);          // B[K, j=lm]
        b0.y = __builtin_nontemporal_load(kr + CCH + lm);
        b1.x = __builtin_nontemporal_load(kr + 16 + lm);     // B[K, j=16+lm]
        b1.y = __builtin_nontemporal_load(kr + CCH + 16 + lm);
        c00 = __builtin_amdgcn_wmma_f32_16x16x4_f32(false, a0, false, b0, (short)0, c00, false, false);
        c01 = __builtin_amdgcn_wmma_f32_16x16x4_f32(false, a0, false, b1, (short)0, c01, false, false);
        c10 = __builtin_amdgcn_wmma_f32_16x16x4_f32(false, a1, false, b0, (short)0, c10, false, false);
        c11 = __builtin_amdgcn_wmma_f32_16x16x4_f32(false, a1, false, b1, (short)0, c11, false, false);
    }

    // Deterministic cross-wave reduction through LDS (no float atomics).
    __shared__ float smem[K1_WAVES * CCH * CCH];   // 32 KB of the 320 KB WGP LDS
    const int rowadd = (lane < 16) ? 0 : 8;        // C/D layout: vgpr r -> M = r + 8*(lane>=16)
    float* sw = smem + wave * (CCH * CCH);
#pragma unroll
    for (int r = 0; r < 8; ++r) {
        const int i0 = r + rowadd;
        sw[(i0     ) * CCH + lm     ] = c00[r];
        sw[(i0     ) * CCH + 16 + lm] = c01[r];
        sw[(i0 + 16) * CCH + lm     ] = c10[r];
        sw[(i0 + 16) * CCH + 16 + lm] = c11[r];
    }
    __syncthreads();

    float* outp = partial + (long)(b * K1_WGS + wg) * (CCH * CCH);
    for (int e = threadIdx.x * 4; e < threadIdx.x * 4 + 4; ++e) {
        float s = 0.f;
        for (int w = 0; w < K1_WAVES; ++w) s += smem[w * (CCH * CCH) + e];
        outp[e] = s;
    }
}

// -------- Pass 2: deterministic partial reduction + row softmax --------------
__global__ __launch_bounds__(256) void softmax_kernel(const float* __restrict__ partial,
                                                      float* __restrict__ wfin) {
    const int b = blockIdx.x;
    const int t = threadIdx.x;
    __shared__ float g[CCH * CCH];
    for (int e = t * 4; e < t * 4 + 4; ++e) {
        float s = 0.f;
        const float* p = partial + (long)b * K1_WGS * (CCH * CCH) + e;
        for (int w = 0; w < K1_WGS; ++w) s += p[(long)w * (CCH * CCH)];
        g[e] = s;
    }
    __syncthreads();
    if (t < CCH) {   // one thread per row i, softmax over j
        float m = -3.4e38f;
#pragma unroll
        for (int j = 0; j < CCH; ++j) m = fmaxf(m, g[t * CCH + j]);
        float ex[CCH];
        float s = 0.f;
#pragma unroll
        for (int j = 0; j < CCH; ++j) { ex[j] = __expf(g[t * CCH + j] - m); s += ex[j]; }
        const float inv = 1.f / s;
#pragma unroll
        for (int j = 0; j < CCH; ++j) wfin[(long)b * (CCH * CCH) + t * CCH + j] = ex[j] * inv;
    }
}

// -------- Pass 3: out[n,i] = q[n,i] + sum_j q[n,j] * w[i,j] ------------------
__global__ __launch_bounds__(256) void apply_kernel(const float* __restrict__ q,
                                                    const float* __restrict__ wfin,
                                                    float* __restrict__ out) {
    const int wg   = blockIdx.x % K3_WGS;
    const int b    = blockIdx.x / K3_WGS;
    const int wave = threadIdx.x >> 5;
    const int lane = threadIdx.x & 31;
    const int lm   = lane & 15;
    const int kb   = (lane < 16) ? 0 : 2;

    // Preload B = w^T in WMMA B layout; K-step kk covers j = 4*kk .. 4*kk+3.
    // Lane holds B[K, n=i_base+lm] = w[i_base+lm][4*kk + kb + {0,1}] (contiguous pair).
    const float* wb = wfin + (long)b * (CCH * CCH);
    v2f Bw0[8], Bw1[8];
#pragma unroll
    for (int kk = 0; kk < 8; ++kk) {
        const float* w0 = wb + lm * CCH + 4 * kk + kb;
        const float* w1 = wb + (16 + lm) * CCH + 4 * kk + kb;
        Bw0[kk].x = w0[0];  Bw0[kk].y = w0[1];
        Bw1[kk].x = w1[0];  Bw1[kk].y = w1[1];
    }

    const long n0 = (long)wg * (8 * K3_ROWS) + (long)wave * K3_ROWS;
    const float* qb = q   + ((long)b * NSP + n0) * CCH;
    float*       ob = out + ((long)b * NSP + n0) * CCH;
    const int rowadd = (lane < 16) ? 0 : 8;

    for (int it = 0; it < K3_ROWS; it += 16) {
        // A[m=lm, K] = q[row it+lm][4*kk + kb + {0,1}] (contiguous float2 per lane)
        v2f A[8];
#pragma unroll
        for (int kk = 0; kk < 8; ++kk) {
            const v2f* qr = (const v2f*)(qb + (long)(it + lm) * CCH + 4 * kk + kb);
            A[kk] = *qr;
        }
        v8f c0 = {}, c1 = {};
#pragma unroll
        for (int kk = 0; kk < 8; ++kk) {
            c0 = __builtin_amdgcn_wmma_f32_16x16x4_f32(false, A[kk], false, Bw0[kk], (short)0, c0, false, false);
            c1 = __builtin_amdgcn_wmma_f32_16x16x4_f32(false, A[kk], false, Bw1[kk], (short)0, c1, false, false);
        }
#pragma unroll
        for (int r = 0; r < 8; ++r) {
            const long row = it + r + rowadd;
            __builtin_nontemporal_store(c0[r] + qb[row * CCH + lm],      ob + row * CCH + lm);
            __builtin_nontemporal_store(c1[r] + qb[row * CCH + 16 + lm], ob + row * CCH + 16 + lm);
        }
    }
}

extern "C" void kernel_launch(void* const* d_in, const int* in_sizes, int n_in,
                              void* d_out, int out_size, void* d_ws, size_t ws_size,
                              hipStream_t stream) {
    const float* q = (const float*)d_in[0];
    const float* k = (const float*)d_in[1];
    float* out = (float*)d_out;
    const int B = in_sizes[0] / (NSP * CCH);   // = 4

    float* partial = (float*)d_ws;                           // B*K1_WGS*1024 floats (2 MB)
    float* wfin    = partial + (long)B * K1_WGS * CCH * CCH; // B*1024 floats (16 KB)

    gram_kernel   <<<dim3(B * K1_WGS), dim3(256), 0, stream>>>(q, k, partial);
    softmax_kernel<<<dim3(B),          dim3(256), 0, stream>>>(partial, wfin);
    apply_kernel  <<<dim3(B * K3_WGS), dim3(256), 0, stream>>>(q, wfin, out);
}